// PrototypeMemory_10144712753746
// MI455X (gfx1250) — compile-verified
//
#include <hip/hip_runtime.h>

typedef __attribute__((ext_vector_type(2))) float v2f;
typedef __attribute__((ext_vector_type(8))) float v8f;

#define N_PTS 262144
#define DIM   128
#define KCLUS 512
#define ITERS 5
#define EPSN  1e-12f

// ---------------------------------------------------------------------------
// Per-wave helper: butterfly-reduce (value,index) across each 16-lane group,
// store assignment, and accumulate this wave's rows into cluster sums/counts.
// After the xor butterfly every lane in a 16-lane group holds the winner, so
// all 16 lanes cooperate on the 128-dim atomic accumulation (8 dims/lane).
// ---------------------------------------------------------------------------
__device__ __forceinline__ void reduce_and_accumulate(
    float bv[8], int bi[8], int row16base, int half, int lq,
    const float* __restrict__ E, int* __restrict__ assign_out,
    float* __restrict__ sums, float* __restrict__ counts)
{
#pragma unroll
  for (int r = 0; r < 8; ++r) {
    float v = bv[r];
    int   i = bi[r];
#pragma unroll
    for (int off = 8; off >= 1; off >>= 1) {
      float ov = __shfl_xor(v, off, 16);
      int   oi = __shfl_xor(i, off, 16);
      if (ov > v || (ov == v && oi < i)) { v = ov; i = oi; }
    }
    const int row = row16base + r + 8 * half;   // lanes 0-15: M=r, lanes 16-31: M=r+8
    if (lq == 0) assign_out[row] = i;
    const float* erow = E    + (size_t)row * DIM + lq * 8;
    float*       srow = sums + (size_t)i   * DIM + lq * 8;
#pragma unroll
    for (int q = 0; q < 8; ++q) {
      unsafeAtomicAdd(&srow[q], erow[q]);     // global_atomic_add_f32 (no return)
    }
    if (lq == 0) unsafeAtomicAdd(&counts[i], 1.0f);
  }
}

// ---------------------------------------------------------------------------
// E-step + scatter M-step. Each wave: 32 rows (two 16x16xK WMMA row tiles),
// full K=512 columns, D=128 accumulated as 32 chunks of the fp32 16x16x4 WMMA.
// ---------------------------------------------------------------------------
__global__ __launch_bounds__(256) void kmeans_assign_kernel(
    const float* __restrict__ E, const float* __restrict__ C,
    int* __restrict__ assign_out, float* __restrict__ sums,
    float* __restrict__ counts)
{
  const int wave  = threadIdx.x >> 5;
  const int lane  = threadIdx.x & 31;
  const int half  = lane >> 4;        // 0: K=d0,d0+1   1: K=d0+2,d0+3
  const int lq    = lane & 15;        // row (A) / column (B,C) within tile
  const int rbase = (blockIdx.x * 8 + wave) * 32;

  // --- Preload A fragments for both row tiles (entire D=128 reduction) ---
  // 16x4 fp32 A layout: lane m (0-15) -> A[m][d0 + 0,1], lane m+16 -> A[m][d0 + 2,3]
  v2f a0[32], a1[32];
  const float* e0 = E + (size_t)(rbase + lq) * DIM + half * 2;
  const float* e1 = e0 + 16 * DIM;
#pragma unroll
  for (int kc = 0; kc < 32; ++kc) {
    a0[kc] = *reinterpret_cast<const v2f*>(e0 + kc * 4);
    a1[kc] = *reinterpret_cast<const v2f*>(e1 + kc * 4);
  }

  float bv0[8], bv1[8];
  int   bi0[8], bi1[8];
#pragma unroll
  for (int r = 0; r < 8; ++r) {
    bv0[r] = -3.402823466e38f; bv1[r] = -3.402823466e38f;
    bi0[r] = 0;                bi1[r] = 0;
  }

  // --- Sweep the 32 column tiles of 16 centroids each ---
  for (int j = 0; j < 32; ++j) {
    // 4x16 fp32 B layout mirrors A: lane n -> B[0,1][n], lane n+16 -> B[2,3][n]
    // B[kd][n] = centroids[j*16+n][d0+kd] -> per-lane contiguous float2 loads.
    const float* cb = C + (size_t)(j * 16 + lq) * DIM + half * 2;
    v8f acc0 = {};
    v8f acc1 = {};
#pragma unroll
    for (int kc = 0; kc < 32; ++kc) {
      v2f b = *reinterpret_cast<const v2f*>(cb + kc * 4);
      acc0 = __builtin_amdgcn_wmma_f32_16x16x4_f32(false, a0[kc], false, b,
                                                   (short)0, acc0, false, false);
      acc1 = __builtin_amdgcn_wmma_f32_16x16x4_f32(false, a1[kc], false, b,
                                                   (short)0, acc1, false, false);
    }
#pragma unroll
    for (int r = 0; r < 8; ++r) {
      const int col = j * 16 + lq;          // C/D layout: N = lane % 16
      const float v0 = acc0[r];
      if (v0 > bv0[r]) { bv0[r] = v0; bi0[r] = col; }
      const float v1 = acc1[r];
      if (v1 > bv1[r]) { bv1[r] = v1; bi1[r] = col; }
    }
  }

  reduce_and_accumulate(bv0, bi0, rbase,      half, lq, E, assign_out, sums, counts);
  reduce_and_accumulate(bv1, bi1, rbase + 16, half, lq, E, assign_out, sums, counts);
}

// ---------------------------------------------------------------------------
// M-step finalize: mean (keep old centroid when empty), then L2-normalize.
// One block per cluster, one thread per dim.
// ---------------------------------------------------------------------------
__global__ __launch_bounds__(128) void kmeans_update_kernel(
    const float* __restrict__ sums, const float* __restrict__ counts,
    const float* __restrict__ oldC, float* __restrict__ newC,
    float* __restrict__ outC)
{
  const int k    = blockIdx.x;
  const int d    = threadIdx.x;
  const int lane = d & 31;
  const int wv   = d >> 5;

  const float cnt = counts[k];
  const size_t idx = (size_t)k * DIM + d;
  float c = (cnt > 0.0f) ? (sums[idx] / fmaxf(cnt, 1.0f)) : oldC[idx];

  // block reduction of c*c over 128 threads (4 waves)
  float s = c * c;
#pragma unroll
  for (int off = 16; off >= 1; off >>= 1) s += __shfl_xor(s, off, 32);

  __shared__ float red[4];
  if (lane == 0) red[wv] = s;
  __syncthreads();
  const float tot  = red[0] + red[1] + red[2] + red[3];
  const float norm = fmaxf(sqrtf(tot), EPSN);

  const float o = c / norm;
  newC[idx] = o;
  if (outC) outC[idx] = o;
}

// ---------------------------------------------------------------------------
extern "C" void kernel_launch(void* const* d_in, const int* in_sizes, int n_in,
                              void* d_out, int out_size, void* d_ws, size_t ws_size,
                              hipStream_t stream) {
  const float* E  = (const float*)d_in[0];   // [N, D]
  const float* C0 = (const float*)d_in[1];   // [K, D]
  // d_in[2] = num_iters (static 5 in reference setup)

  int*   assign_out = (int*)d_out;                    // first N int32
  float* cent_out   = (float*)d_out + N_PTS;          // then K*D float32

  float* ws_cent   = (float*)d_ws;                    // K*D current centroids
  float* ws_sums   = ws_cent + (size_t)KCLUS * DIM;   // K*D cluster sums
  float* ws_counts = ws_sums + (size_t)KCLUS * DIM;   // K counts

  for (int t = 0; t < ITERS; ++t) {
    const float* cur = (t == 0) ? C0 : ws_cent;
    hipMemsetAsync(ws_sums, 0, ((size_t)KCLUS * DIM + KCLUS) * sizeof(float), stream);
    kmeans_assign_kernel<<<N_PTS / 256, 256, 0, stream>>>(
        E, cur, assign_out, ws_sums, ws_counts);
    kmeans_update_kernel<<<KCLUS, 128, 0, stream>>>(
        ws_sums, ws_counts, cur, ws_cent, (t == ITERS - 1) ? cent_out : nullptr);
  }
}